// SphereMaxPool2D_21543555957402
// MI455X (gfx1250) — compile-verified
//
#include <hip/hip_runtime.h>
#include <math.h>

// SphereMaxPool2D for MI455X (gfx1250).
// Input fits in 192MB L2 -> gather-heavy bilinear taps are L2 hits; HBM floor
// = 128MB read + 32MB write ~= 7us @ 23.3TB/s. Coord trig is hoisted to a tiny
// pre-pass; per-block tap tables are staged into LDS with CDNA5 async
// global->LDS loads (ASYNCcnt) and reused across 8 channel planes.

#define HH   256
#define WW   512
#define HS   128
#define WSZ  256
#define NB   8
#define NC   32
#define BC   (NB*NC)
#define STR  2
#define NTAP 9
#define WTILE 32
#define PCHUNK 8

struct TapIdx { int r0, r1, c0, c1; };   // 16 B
struct TapW   { float d1, d2;      };    // 8 B

__device__ __forceinline__ int wrap_idx(float a, int m) {
    // matches jnp.mod(float, m).astype(int32)
    float q = fmodf(a, (float)m);
    if (q < 0.0f) q += (float)m;
    int i = (int)q;
    return (i >= m) ? 0 : i;   // guard fp edge
}

__global__ void sphere_coords_kernel(TapIdx* __restrict__ idx_tab,
                                     TapW*   __restrict__ w_tab) {
    int k = blockIdx.x * blockDim.x + threadIdx.x;
    if (k >= HS * WSZ * NTAP) return;
    int tap = k % NTAP;
    int wo  = (k / NTAP) % WSZ;
    int ho  =  k / (NTAP * WSZ);

    const double PI = 3.14159265358979323846;
    double img_r = (double)(STR * ho);
    double img_c = (double)(STR * wo);
    double nr, nc;

    if (tap == 4) {                       // center tap is overridden to exact coords
        nr = img_r; nc = img_c;
    } else {
        // stencil constants (f64, cheap, same for all threads)
        double dphi = PI / HH, dth = 2.0 * PI / WW;
        double tt  = tan(dth), tp = tan(dphi), cpv = tp / cos(dth);
        int row = tap / 3, col = tap % 3;
        double xf = (col == 0) ? -tt : (col == 2) ? tt : ((row == 1) ? 1.0 : 0.0);
        double yf = (col == 1) ? ((row == 0) ? tp  : (row == 1) ? 1.0 : -tp)
                               : ((row == 0) ? cpv : (row == 1) ? 0.0 : -cpv);
        double rho = sqrt(xf * xf + yf * yf);
        float  v   = (float)atan(rho);
        float  sv  = sinf(v), cv = cosf(v);
        float  phi   = (float)(-((img_r + 0.5) / HH * PI - PI / 2.0));
        float  theta = (float)((img_c + 0.5) / WW * 2.0 * PI - PI);
        float  sp = sinf(phi), cp = cosf(phi);
        float  frho = (float)rho, fx = (float)xf, fy = (float)yf;
        float  new_phi = asinf(cv * sp + fy * sv * cp / frho);
        float  dtheta  = atanf(fx * sv / (frho * cp * cv - fy * sp * sv));
        nr = ((float)(PI / 2.0) - new_phi) * ((float)HH / (float)PI) - 0.5f;
        nc = (theta + dtheta + (float)PI) * ((float)WW / (float)(2.0 * PI)) - 0.5f;
        nc = fmod(nc + (double)WW, (double)WW);
    }

    float r = (float)nr, c = (float)nc;
    float rf = floorf(r), rc = ceilf(r);
    float cf = floorf(c), cc = ceilf(c);

    TapIdx ti;
    ti.r0 = wrap_idx(rf, HH); ti.r1 = wrap_idx(rc, HH);
    ti.c0 = wrap_idx(cf, WW); ti.c1 = wrap_idx(cc, WW);
    idx_tab[k] = ti;
    TapW tw; tw.d1 = c - cf; tw.d2 = r - rf;
    w_tab[k] = tw;
}

__global__ __launch_bounds__(WTILE * PCHUNK)
void sphere_pool_kernel(const float* __restrict__ x,
                        const TapIdx* __restrict__ idx_tab,
                        const TapW*   __restrict__ w_tab,
                        float* __restrict__ out) {
    __shared__ TapIdx s_idx[WTILE * NTAP];
    __shared__ TapW   s_w  [WTILE * NTAP];

    const int tid = threadIdx.x;
    const int wo0 = blockIdx.x * WTILE;
    const int ho  = blockIdx.y;
    const int p0  = blockIdx.z * PCHUNK;
    const int k0  = (ho * WSZ + wo0) * NTAP;

    // ---- Phase 1: CDNA5 async global->LDS staging of the tap tables ----
    {
        const void* gidx = (const void*)(idx_tab + k0);
        const void* gw   = (const void*)(w_tab  + k0);
        for (int e = tid; e < WTILE * NTAP; e += WTILE * PCHUNK) {
            unsigned lds_i = (unsigned)(size_t)(&s_idx[e]);   // addr[31:0] = LDS offset
            unsigned off_i = (unsigned)(e * (int)sizeof(TapIdx));
            asm volatile("global_load_async_to_lds_b128 %0, %1, %2 offset:0"
                         :: "v"(lds_i), "v"(off_i), "s"(gidx) : "memory");
            unsigned lds_w = (unsigned)(size_t)(&s_w[e]);
            unsigned off_w = (unsigned)(e * (int)sizeof(TapW));
            asm volatile("global_load_async_to_lds_b64 %0, %1, %2 offset:0"
                         :: "v"(lds_w), "v"(off_w), "s"(gw) : "memory");
        }
        asm volatile("s_wait_asynccnt 0" ::: "memory");
    }
    __syncthreads();

    // ---- Phase 2: 9-tap bilinear + max, 8 planes per block reuse the LDS table
    const int wl    = tid & (WTILE - 1);
    const int p     = tid / WTILE;
    const int plane = p0 + p;
    const float* __restrict__ xp = x + (size_t)plane * (HH * WW);

    float best = -INFINITY;
#pragma unroll
    for (int tap = 0; tap < NTAP; ++tap) {
        TapIdx ti = s_idx[wl * NTAP + tap];
        TapW   tw = s_w  [wl * NTAP + tap];
        float f00 = xp[ti.r0 * WW + ti.c0];
        float f10 = xp[ti.r0 * WW + ti.c1];
        float f01 = xp[ti.r1 * WW + ti.c0];
        float f11 = xp[ti.r1 * WW + ti.c1];
        float fx1 = f00 + tw.d1 * (f10 - f00);
        float fx2 = f01 + tw.d1 * (f11 - f01);
        float s   = fx1 + tw.d2 * (fx2 - fx1);
        best = fmaxf(best, s);
    }
    out[(size_t)plane * (HS * WSZ) + (size_t)ho * WSZ + wo0 + wl] = best;
}

extern "C" void kernel_launch(void* const* d_in, const int* in_sizes, int n_in,
                              void* d_out, int out_size, void* d_ws, size_t ws_size,
                              hipStream_t stream) {
    const float* x   = (const float*)d_in[0];
    float*       out = (float*)d_out;

    const size_t n_taps   = (size_t)HS * WSZ * NTAP;           // 294,912
    TapIdx* idx_tab = (TapIdx*)d_ws;                            // 4.72 MB
    TapW*   w_tab   = (TapW*)((char*)d_ws + n_taps * sizeof(TapIdx)); // +2.36 MB

    {
        int n = (int)n_taps;
        int blocks = (n + 255) / 256;
        sphere_coords_kernel<<<blocks, 256, 0, stream>>>(idx_tab, w_tab);
    }
    {
        dim3 grid(WSZ / WTILE, HS, BC / PCHUNK);                // (8,128,32)
        sphere_pool_kernel<<<grid, WTILE * PCHUNK, 0, stream>>>(x, idx_tab, w_tab, out);
    }
}